// DRL4SSP_41712722379132
// MI455X (gfx1250) — compile-verified
//
#include <hip/hip_runtime.h>

typedef __attribute__((ext_vector_type(2))) float v2f;
typedef __attribute__((ext_vector_type(8))) float v8f;

// ---------------- workspace layout (float offsets) ----------------
// All B-operand weight matrices are stored K-PAIR-PACKED: layout [K/2][N][2],
// element (p,n) = { W^T[2p][n], W^T[2p+1][n] } so one b64 load = one WMMA B operand.
#define OFF_WIHT   0u          // 128*384  W_ih^T  pair-packed
#define OFF_WHHT   49152u      // 128*384  W_hh^T  pair-packed
#define OFF_W1HT   98304u      // 128*128  w1h^T   pair-packed (ww1[:,2H:])^T
#define OFF_W2DT   114688u     // 128*128  w2d^T   pair-packed (ww2[:,H:2H])^T
#define OFF_WDECT  131072u     // 8*128    W_dec^T pair-packed
#define OFF_WST    132096u     // 8*128    W_s^T   pair-packed
#define OFF_M1S    133120u     // 128*8    ww1[:,:H] @ W_s
#define OFF_M1D    134144u     // 128*4    ww1[:,H:2H] @ W_d
#define OFF_M2S    134656u     // 128*8    ww2[:,:H] @ W_s
#define OFF_M2D    135680u     // 128*4    ww2[:,2H:] @ W_d
#define OFF_B1V    136192u     // 128      bias1
#define OFF_B2V    136320u     // 128      bias2
#define OFF_MASK   136448u     // 64*128   mask0
#define OFF_BASE1  144640u     // 64*128*128
#define OFF_BASE2  1193216u    // 64*128*128
// total = 2241792 floats ~= 8.97 MB

#define BPB 8   // batches per block (rec kernel); 8 blocks total

// ---------------- CDNA5 WMMA f32 16x16x4 wrapper ----------------
static __device__ __forceinline__ v8f wmma4(v2f a, v2f b, v8f c) {
  // D = A(16x4 f32) * B(4x16 f32) + C(16x16 f32)
  return __builtin_amdgcn_wmma_f32_16x16x4_f32(
      /*neg_a=*/false, a, /*neg_b=*/false, b,
      /*c_mod=*/(short)0, c, /*reuse_a=*/false, /*reuse_b=*/false);
}

static __device__ __forceinline__ float fast_sig(float x) {
  return __fdividef(1.0f, 1.0f + __expf(-x));
}

#if __has_builtin(__builtin_amdgcn_tanhf)
static __device__ __forceinline__ float fast_tanh(float x) {
  return __builtin_amdgcn_tanhf(x);   // v_tanh_f32 on gfx1250
}
#else
static __device__ __forceinline__ float fast_tanh(float x) {
  float e = __expf(-2.0f * fabsf(x));
  float t = __fdividef(1.0f - e, 1.0f + e);
  return x < 0.0f ? -t : t;
}
#endif

static __device__ __forceinline__ float wred_sum(float v) {
  #pragma unroll
  for (int off = 16; off > 0; off >>= 1) v += __shfl_xor(v, off, 32);
  return v;
}

// ---------------- kernel 0: weight repack (pair-interleave) + folded matrices ----------------
__global__ void __launch_bounds__(256)
prep_kernel(const float* __restrict__ Ws, const float* __restrict__ bs,
            const float* __restrict__ Wd, const float* __restrict__ bd,
            const float* __restrict__ Wdec,
            const float* __restrict__ ww1, const float* __restrict__ ww2,
            const float* __restrict__ Wih, const float* __restrict__ Whh,
            float* __restrict__ ws)
{
  const int tid = threadIdx.x;
  // W_ih^T / W_hh^T pair-packed: (p, n) -> { W[n][2p], W[n][2p+1] }
  for (int idx = tid; idx < 64 * 384; idx += 256) {
    int p = idx / 384, n = idx % 384;
    ws[OFF_WIHT + 2 * idx]     = Wih[n * 128 + 2 * p];
    ws[OFF_WIHT + 2 * idx + 1] = Wih[n * 128 + 2 * p + 1];
    ws[OFF_WHHT + 2 * idx]     = Whh[n * 128 + 2 * p];
    ws[OFF_WHHT + 2 * idx + 1] = Whh[n * 128 + 2 * p + 1];
  }
  // w1h^T / w2d^T pair-packed
  for (int idx = tid; idx < 64 * 128; idx += 256) {
    int p = idx >> 7, n = idx & 127;
    ws[OFF_W1HT + 2 * idx]     = ww1[n * 384 + 256 + 2 * p];
    ws[OFF_W1HT + 2 * idx + 1] = ww1[n * 384 + 256 + 2 * p + 1];
    ws[OFF_W2DT + 2 * idx]     = ww2[n * 384 + 128 + 2 * p];
    ws[OFF_W2DT + 2 * idx + 1] = ww2[n * 384 + 128 + 2 * p + 1];
  }
  // W_dec^T / W_s^T pair-packed (K=8 -> 4 pair rows)
  for (int idx = tid; idx < 4 * 128; idx += 256) {
    int p = idx >> 7, n = idx & 127;
    ws[OFF_WDECT + 2 * idx]     = Wdec[n * 8 + 2 * p];
    ws[OFF_WDECT + 2 * idx + 1] = Wdec[n * 8 + 2 * p + 1];
    ws[OFF_WST + 2 * idx]       = Ws[n * 8 + 2 * p];
    ws[OFF_WST + 2 * idx + 1]   = Ws[n * 8 + 2 * p + 1];
  }
  for (int idx = tid; idx < 1024; idx += 256) {   // M1s, M2s (128x8)
    int h = idx >> 3, i = idx & 7;
    float s1 = 0.f, s2 = 0.f;
    for (int k = 0; k < 128; ++k) {
      float w = Ws[k * 8 + i];
      s1 += ww1[h * 384 + k] * w;
      s2 += ww2[h * 384 + k] * w;
    }
    ws[OFF_M1S + idx] = s1;
    ws[OFF_M2S + idx] = s2;
  }
  for (int idx = tid; idx < 512; idx += 256) {    // M1d, M2d (128x4)
    int h = idx >> 2, j = idx & 3;
    float s1 = 0.f, s2 = 0.f;
    for (int k = 0; k < 128; ++k) {
      float w = Wd[k * 4 + j];
      s1 += ww1[h * 384 + 128 + k] * w;
      s2 += ww2[h * 384 + 256 + k] * w;
    }
    ws[OFF_M1D + idx] = s1;
    ws[OFF_M2D + idx] = s2;
  }
  for (int h = tid; h < 128; h += 256) {          // bias1, bias2
    float s1 = 0.f, s2 = 0.f;
    for (int k = 0; k < 128; ++k) {
      s1 += ww1[h * 384 + k] * bs[k] + ww1[h * 384 + 128 + k] * bd[k];
      s2 += ww2[h * 384 + k] * bs[k] + ww2[h * 384 + 256 + k] * bd[k];
    }
    ws[OFF_B1V + h] = s1;
    ws[OFF_B2V + h] = s2;
  }
}

// ---------------- kernel 1: base1/base2 (K=12 folded contraction) + mask0 ----------------
__global__ void __launch_bounds__(256)
base_kernel(const float* __restrict__ st, const float* __restrict__ dy,
            float* __restrict__ ws)
{
  const int b = blockIdx.x;
  const int tid = threadIdx.x;
  const float* M1s = ws + OFF_M1S; const float* M1d = ws + OFF_M1D;
  const float* M2s = ws + OFF_M2S; const float* M2d = ws + OFF_M2D;
  const float* b1v = ws + OFF_B1V; const float* b2v = ws + OFF_B2V;
  float* base1 = ws + OFF_BASE1 + (size_t)b * 128 * 128;
  float* base2 = ws + OFF_BASE2 + (size_t)b * 128 * 128;
  for (int idx = tid; idx < 128 * 128; idx += 256) {
    int h = idx >> 7, s = idx & 127;
    float v1 = b1v[h], v2 = b2v[h];
    #pragma unroll
    for (int i = 0; i < 8; ++i) {
      float x = st[((size_t)b * 8 + i) * 128 + s];
      v1 += M1s[h * 8 + i] * x;
      v2 += M2s[h * 8 + i] * x;
    }
    #pragma unroll
    for (int j = 0; j < 4; ++j) {
      float x = dy[((size_t)b * 4 + j) * 128 + s];
      v1 += M1d[h * 4 + j] * x;
      v2 += M2d[h * 4 + j] * x;
    }
    base1[idx] = v1;
    base2[idx] = v2;
  }
  if (tid < 128) {
    int s = tid;
    float d0 = dy[((size_t)b * 4) * 128 + s];
    float m = (d0 != 0.f) ? 0.f : 1.f;
    if (s == 0) m = 0.f;
    ws[OFF_MASK + b * 128 + s] = m;
  }
}

// ---------------- kernel 2: persistent recurrent decode loop ----------------
__global__ void __launch_bounds__(256)
rec_kernel(const float* __restrict__ st, const float* __restrict__ vv1,
           const float* __restrict__ vv2, const float* __restrict__ bs,
           const float* __restrict__ bdec, const float* __restrict__ bih,
           const float* __restrict__ bhh, const float* __restrict__ ws,
           float* __restrict__ out)
{
  __shared__ __align__(16) float sm_hA[16 * 128];
  __shared__ __align__(16) float sm_hB[16 * 128];
  __shared__ __align__(16) float sm_x[16 * 128];    // dec_h, later dytext
  __shared__ __align__(16) float sm_q[16 * 128];    // q1, later q2
  __shared__ __align__(16) float sm_dec[16 * 8];    // dec_in
  __shared__ __align__(16) float sm_u[16 * 8];      // u = static @ a1
  __shared__ __align__(16) float sm_mask[16 * 128];

  const int tid  = threadIdx.x;
  const int wave = tid >> 5;
  const int lane = tid & 31;
  const int half = lane >> 4;
  const int lm   = lane & 15;
  const int b0   = blockIdx.x * BPB;

  const float* base1 = ws + OFF_BASE1;
  const float* base2 = ws + OFF_BASE2;

  for (int i = tid; i < 16 * 128; i += 256) {
    sm_hA[i] = 0.f; sm_hB[i] = 0.f; sm_x[i] = 0.f; sm_q[i] = 0.f;
    int bi = i >> 7, s = i & 127;
    sm_mask[i] = (bi < BPB) ? ws[OFF_MASK + (b0 + bi) * 128 + s] : 0.f;
  }
  for (int i = tid; i < 16 * 8; i += 256) { sm_dec[i] = 0.f; sm_u[i] = 0.f; }
  __syncthreads();

  float* curh = sm_hA;
  float* nxth = sm_hB;
  const int nw  = wave * 16;
  const int col = nw + lm;

  // per-lane strength-reduced base pointers (pair row = k0/2 + half)
  const float* WihB  = ws + OFF_WIHT  + ((size_t)half * 384 + col) * 2;
  const float* WhhB  = ws + OFF_WHHT  + ((size_t)half * 384 + col) * 2;
  const float* w1hB  = ws + OFF_W1HT  + ((size_t)half * 128 + col) * 2;
  const float* w2dB  = ws + OFF_W2DT  + ((size_t)half * 128 + col) * 2;
  const float* WdecB = ws + OFF_WDECT + ((size_t)half * 128 + col) * 2;
  const float* WsB   = ws + OFF_WST   + ((size_t)half * 128 + col) * 2;
  const int aoff = 2 * half;   // A-operand column offset within a k-group

  for (int t = 0; t < 127; ++t) {
    // ---- stage 1: dec_h = dec_in(16x8) @ Wdec^T + b_dec -> sm_x ----
    {
      v8f acc = {0.f,0.f,0.f,0.f,0.f,0.f,0.f,0.f};
      const float* Ad = &sm_dec[lm * 8 + aoff];
      const float* Bd = WdecB;
      #pragma unroll
      for (int k0 = 0; k0 < 8; k0 += 4) {
        acc = wmma4(*(const v2f*)Ad, *(const v2f*)Bd, acc);
        Ad += 4; Bd += 512;                 // 2 pair-rows * 128 * 2
      }
      float bb = bdec[col];
      #pragma unroll
      for (int r = 0; r < 8; ++r) sm_x[(r + 8 * half) * 128 + col] = acc[r] + bb;
    }
    __syncthreads();

    // ---- stage 2: GRU; wave owns (r,z,n) column-triple, gates fuse in regs ----
    {
      v8f gir = {0.f,0.f,0.f,0.f,0.f,0.f,0.f,0.f}, giz = gir, gin = gir;
      v8f ghr = gir, ghz = gir, ghn = gir;
      const float* Ax = &sm_x[lm * 128 + aoff];
      const float* Ah = &curh[lm * 128 + aoff];
      const float* Bi = WihB;
      const float* Bh = WhhB;
      #pragma unroll 4
      for (int k0 = 0; k0 < 128; k0 += 4) {
        v2f ax = *(const v2f*)Ax;
        v2f ah = *(const v2f*)Ah;
        gir = wmma4(ax, *(const v2f*)(Bi),       gir);
        giz = wmma4(ax, *(const v2f*)(Bi + 256), giz);   // +128 cols * 2
        gin = wmma4(ax, *(const v2f*)(Bi + 512), gin);   // +256 cols * 2
        ghr = wmma4(ah, *(const v2f*)(Bh),       ghr);
        ghz = wmma4(ah, *(const v2f*)(Bh + 256), ghz);
        ghn = wmma4(ah, *(const v2f*)(Bh + 512), ghn);
        Ax += 4; Ah += 4;
        Bi += 1536; Bh += 1536;             // 2 pair-rows * 384 * 2
      }
      float bir = bih[col], biz = bih[col + 128], bin = bih[col + 256];
      float bhr = bhh[col], bhz = bhh[col + 128], bhn = bhh[col + 256];
      #pragma unroll
      for (int r = 0; r < 8; ++r) {
        int m = r + 8 * half;
        float hp = curh[m * 128 + col];
        float rg = fast_sig(gir[r] + bir + ghr[r] + bhr);
        float zg = fast_sig(giz[r] + biz + ghz[r] + bhz);
        float ng = fast_tanh(gin[r] + bin + rg * (ghn[r] + bhn));
        nxth[m * 128 + col] = (1.f - zg) * ng + zg * hp;
      }
    }
    __syncthreads();

    // ---- stage 3: q1 = h_new @ w1h^T -> sm_q ----
    {
      v8f acc = {0.f,0.f,0.f,0.f,0.f,0.f,0.f,0.f};
      const float* Aq = &nxth[lm * 128 + aoff];
      const float* Bq = w1hB;
      #pragma unroll 4
      for (int k0 = 0; k0 < 128; k0 += 4) {
        acc = wmma4(*(const v2f*)Aq, *(const v2f*)Bq, acc);
        Aq += 4; Bq += 512;                 // 2 pair-rows * 128 * 2
      }
      #pragma unroll
      for (int r = 0; r < 8; ++r) sm_q[(r + 8 * half) * 128 + col] = acc[r];
    }
    __syncthreads();

    // ---- stage 4: attn1 + softmax + u (one batch per wave) ----
    {
      const int bi = wave;
      const int b  = b0 + bi;
      const float* bp = base1 + (size_t)b * 128 * 128 + lane * 4;
      const float* qp = sm_q + bi * 128;
      float a0 = 0.f, a1 = 0.f, a2 = 0.f, a3 = 0.f;
      for (int h = 0; h < 128; h += 4) {
        const float4 q4 = *(const float4*)&qp[h];
        const float4 v4 = *(const float4*)&vv1[h];
        const float4 x0 = *(const float4*)&bp[(h + 0) * 128];
        const float4 x1 = *(const float4*)&bp[(h + 1) * 128];
        const float4 x2 = *(const float4*)&bp[(h + 2) * 128];
        const float4 x3 = *(const float4*)&bp[(h + 3) * 128];
        a0 += v4.x * fast_tanh(x0.x + q4.x) + v4.y * fast_tanh(x1.x + q4.y)
            + v4.z * fast_tanh(x2.x + q4.z) + v4.w * fast_tanh(x3.x + q4.w);
        a1 += v4.x * fast_tanh(x0.y + q4.x) + v4.y * fast_tanh(x1.y + q4.y)
            + v4.z * fast_tanh(x2.y + q4.z) + v4.w * fast_tanh(x3.y + q4.w);
        a2 += v4.x * fast_tanh(x0.z + q4.x) + v4.y * fast_tanh(x1.z + q4.y)
            + v4.z * fast_tanh(x2.z + q4.z) + v4.w * fast_tanh(x3.z + q4.w);
        a3 += v4.x * fast_tanh(x0.w + q4.x) + v4.y * fast_tanh(x1.w + q4.y)
            + v4.z * fast_tanh(x2.w + q4.z) + v4.w * fast_tanh(x3.w + q4.w);
      }
      float mx = fmaxf(fmaxf(a0, a1), fmaxf(a2, a3));
      #pragma unroll
      for (int off = 16; off > 0; off >>= 1) mx = fmaxf(mx, __shfl_xor(mx, off, 32));
      float e0 = __expf(a0 - mx), e1 = __expf(a1 - mx);
      float e2 = __expf(a2 - mx), e3 = __expf(a3 - mx);
      float inv = __fdividef(1.f, wred_sum(e0 + e1 + e2 + e3));
      e0 *= inv; e1 *= inv; e2 *= inv; e3 *= inv;
      #pragma unroll
      for (int i = 0; i < 8; ++i) {
        const float4 sv = *(const float4*)&st[((size_t)b * 8 + i) * 128 + lane * 4];
        float p = wred_sum(e0 * sv.x + e1 * sv.y + e2 * sv.z + e3 * sv.w);
        if (lane == 0) sm_u[bi * 8 + i] = p;
      }
    }
    __syncthreads();

    // ---- stage 5: dytext = u(16x8) @ W_s^T + b_s -> sm_x ----
    {
      v8f acc = {0.f,0.f,0.f,0.f,0.f,0.f,0.f,0.f};
      const float* Au = &sm_u[lm * 8 + aoff];
      const float* Bu = WsB;
      #pragma unroll
      for (int k0 = 0; k0 < 8; k0 += 4) {
        acc = wmma4(*(const v2f*)Au, *(const v2f*)Bu, acc);
        Au += 4; Bu += 512;
      }
      float bb = bs[col];
      #pragma unroll
      for (int r = 0; r < 8; ++r) sm_x[(r + 8 * half) * 128 + col] = acc[r] + bb;
    }
    __syncthreads();

    // ---- stage 6: q2 = dytext @ w2d^T -> sm_q ----
    {
      v8f acc = {0.f,0.f,0.f,0.f,0.f,0.f,0.f,0.f};
      const float* Aq = &sm_x[lm * 128 + aoff];
      const float* Bq = w2dB;
      #pragma unroll 4
      for (int k0 = 0; k0 < 128; k0 += 4) {
        acc = wmma4(*(const v2f*)Aq, *(const v2f*)Bq, acc);
        Aq += 4; Bq += 512;
      }
      #pragma unroll
      for (int r = 0; r < 8; ++r) sm_q[(r + 8 * half) * 128 + col] = acc[r];
    }
    __syncthreads();

    // ---- stage 7: attn2 + masked argmax + logp + state update ----
    {
      const int bi = wave;
      const int b  = b0 + bi;
      const float* bp = base2 + (size_t)b * 128 * 128 + lane * 4;
      const float* qp = sm_q + bi * 128;
      float a0 = 0.f, a1 = 0.f, a2 = 0.f, a3 = 0.f;
      for (int h = 0; h < 128; h += 4) {
        const float4 q4 = *(const float4*)&qp[h];
        const float4 v4 = *(const float4*)&vv2[h];
        const float4 x0 = *(const float4*)&bp[(h + 0) * 128];
        const float4 x1 = *(const float4*)&bp[(h + 1) * 128];
        const float4 x2 = *(const float4*)&bp[(h + 2) * 128];
        const float4 x3 = *(const float4*)&bp[(h + 3) * 128];
        a0 += v4.x * fast_tanh(x0.x + q4.x) + v4.y * fast_tanh(x1.x + q4.y)
            + v4.z * fast_tanh(x2.x + q4.z) + v4.w * fast_tanh(x3.x + q4.w);
        a1 += v4.x * fast_tanh(x0.y + q4.x) + v4.y * fast_tanh(x1.y + q4.y)
            + v4.z * fast_tanh(x2.y + q4.z) + v4.w * fast_tanh(x3.y + q4.w);
        a2 += v4.x * fast_tanh(x0.z + q4.x) + v4.y * fast_tanh(x1.z + q4.y)
            + v4.z * fast_tanh(x2.z + q4.z) + v4.w * fast_tanh(x3.z + q4.w);
        a3 += v4.x * fast_tanh(x0.w + q4.x) + v4.y * fast_tanh(x1.w + q4.y)
            + v4.z * fast_tanh(x2.w + q4.z) + v4.w * fast_tanh(x3.w + q4.w);
      }
      float lg[4] = {a0, a1, a2, a3};
      #pragma unroll
      for (int j = 0; j < 4; ++j) {
        float mk = sm_mask[bi * 128 + lane * 4 + j];
        lg[j] += (mk > 0.f) ? 0.f : -1e30f;
      }
      float bv = lg[0]; int bidx = lane * 4;
      #pragma unroll
      for (int j = 1; j < 4; ++j)
        if (lg[j] > bv) { bv = lg[j]; bidx = lane * 4 + j; }
      #pragma unroll
      for (int off = 16; off > 0; off >>= 1) {
        float ov = __shfl_xor(bv, off, 32);
        int   oi = __shfl_xor(bidx, off, 32);
        if (ov > bv || (ov == bv && oi < bidx)) { bv = ov; bidx = oi; }
      }
      float ss = __expf(lg[0] - bv) + __expf(lg[1] - bv) +
                 __expf(lg[2] - bv) + __expf(lg[3] - bv);
      ss = wred_sum(ss);
      float logp = -__logf(ss);   // logits[ptr] == max
      if (lane == 0) {
        out[(size_t)b * 127 + t] = (float)bidx;                 // tour_idx
        out[(size_t)64 * 127 + (size_t)b * 127 + t] = logp;     // tour_logp
        sm_mask[bi * 128 + bidx] = 0.f;
      }
      if (lane < 8)
        sm_dec[bi * 8 + lane] = st[((size_t)b * 8 + lane) * 128 + bidx];
    }
    __syncthreads();

    float* tmp = curh; curh = nxth; nxth = tmp;
  }
}

extern "C" void kernel_launch(void* const* d_in, const int* in_sizes, int n_in,
                              void* d_out, int out_size, void* d_ws, size_t ws_size,
                              hipStream_t stream) {
  const float* st   = (const float*)d_in[0];
  const float* dy   = (const float*)d_in[1];
  // d_in[2] transition_time: unused by reference
  const float* Ws   = (const float*)d_in[3];
  const float* bs   = (const float*)d_in[4];
  const float* Wd   = (const float*)d_in[5];
  const float* bd   = (const float*)d_in[6];
  const float* Wdec = (const float*)d_in[7];
  const float* bdec = (const float*)d_in[8];
  const float* vv1  = (const float*)d_in[9];
  const float* ww1  = (const float*)d_in[10];
  const float* vv2  = (const float*)d_in[11];
  const float* ww2  = (const float*)d_in[12];
  const float* Wih  = (const float*)d_in[13];
  const float* Whh  = (const float*)d_in[14];
  const float* bih  = (const float*)d_in[15];
  const float* bhh  = (const float*)d_in[16];
  float* ws  = (float*)d_ws;
  float* out = (float*)d_out;

  prep_kernel<<<dim3(1),  dim3(256), 0, stream>>>(Ws, bs, Wd, bd, Wdec, ww1, ww2, Wih, Whh, ws);
  base_kernel<<<dim3(64), dim3(256), 0, stream>>>(st, dy, ws);
  rec_kernel <<<dim3(8),  dim3(256), 0, stream>>>(st, vv1, vv2, bs, bdec, bih, bhh, ws, out);
}